// BoundaryDiceLoss_29746943492528
// MI455X (gfx1250) — compile-verified
//
#include <hip/hip_runtime.h>

#define NCLS   5
#define HDIM   512
#define WDIM   512
#define HW     (HDIM * WDIM)       // 262144
#define BIMG   32
#define BPI    32                  // blocks per image
#define TPB    256
#define PPT    (HW / (BPI * TPB))  // 32 pixels per thread
#define NSTAT  30
#define SLOT   32
#define EPSF   1e-6f

typedef __attribute__((ext_vector_type(2))) float v2f;
typedef __attribute__((ext_vector_type(8))) float v8f;

// ---------------------------------------------------------------------------
// Kernel 1: per-block partial sums of the 30 statistics for one image chunk.
// Streaming, bandwidth-bound: float4/int4 coalesced loads, register accum,
// wave32 shuffle reduction, one 32-float slot per block in d_ws.
// ---------------------------------------------------------------------------
__global__ __launch_bounds__(TPB) void bd_partial(const float* __restrict__ pred,
                                                  const int* __restrict__ tgt,
                                                  float* __restrict__ ws)
{
    const int img  = blockIdx.x / BPI;
    const int blk  = blockIdx.x % BPI;
    const int tid  = threadIdx.x;
    const int base = blk * (TPB * PPT);

    const float* __restrict__ predb = pred + (size_t)img * NCLS * HW;
    const int*   __restrict__ tgtb  = tgt  + (size_t)img * HW;

    float sumSoft[NCLS]  = {0,0,0,0,0};
    float sumSoftW[NCLS] = {0,0,0,0,0};
    float inter[NCLS]    = {0,0,0,0,0};
    float interW[NCLS]   = {0,0,0,0,0};
    float cnt[NCLS]      = {0,0,0,0,0};
    float cntW[NCLS]     = {0,0,0,0,0};

    for (int k = 0; k < PPT / 4; ++k) {
        const int p0 = base + (k * TPB + tid) * 4;   // 4 consecutive pixels, same row
        const int y  = p0 >> 9;
        const int x0 = p0 & (WDIM - 1);

        const int4 t4 = *(const int4*)(tgtb + p0);
        int tt[4] = { t4.x, t4.y, t4.z, t4.w };

        const bool hasUp = (y > 0);
        const bool hasDn = (y < HDIM - 1);
        int up[4] = { tt[0], tt[1], tt[2], tt[3] };   // OOB == same label (pad=1)
        int dn[4] = { tt[0], tt[1], tt[2], tt[3] };
        if (hasUp) { int4 u = *(const int4*)(tgtb + p0 - WDIM); up[0]=u.x; up[1]=u.y; up[2]=u.z; up[3]=u.w; }
        if (hasDn) { int4 d = *(const int4*)(tgtb + p0 + WDIM); dn[0]=d.x; dn[1]=d.y; dn[2]=d.z; dn[3]=d.w; }
        const int lft = (x0 > 0)          ? tgtb[p0 - 1] : tt[0];
        const int rgt = (x0 < WDIM - 4)   ? tgtb[p0 + 4] : tt[3];
        int nbrL[4] = { lft, tt[0], tt[1], tt[2] };
        int nbrR[4] = { tt[1], tt[2], tt[3], rgt };

        float qa[NCLS][4];
#pragma unroll
        for (int c = 0; c < NCLS; ++c) {
            float4 q = *(const float4*)(predb + (size_t)c * HW + p0);
            qa[c][0] = q.x; qa[c][1] = q.y; qa[c][2] = q.z; qa[c][3] = q.w;
        }

#pragma unroll
        for (int j = 0; j < 4; ++j) {
            const int t = tt[j];
            float w = 0.0f;
            if (t != 0) {
                bool bnd = (up[j] != t) | (dn[j] != t) | (nbrL[j] != t) | (nbrR[j] != t);
                w = bnd ? 1.0f : 0.0f;
            }
            const float v0 = qa[0][j], v1 = qa[1][j], v2 = qa[2][j],
                        v3 = qa[3][j], v4 = qa[4][j];
            const float m  = fmaxf(fmaxf(fmaxf(v0, v1), fmaxf(v2, v3)), v4);
            const float e0 = __expf(v0 - m), e1 = __expf(v1 - m), e2 = __expf(v2 - m),
                        e3 = __expf(v3 - m), e4 = __expf(v4 - m);
            const float inv = 1.0f / (e0 + e1 + e2 + e3 + e4);
            const float s[NCLS] = { e0*inv, e1*inv, e2*inv, e3*inv, e4*inv };
#pragma unroll
            for (int c = 0; c < NCLS; ++c) {
                const float sc  = s[c];
                const float ist = (t == c) ? 1.0f : 0.0f;
                sumSoft[c]  += sc;
                sumSoftW[c] += sc * w;
                inter[c]    += sc * ist;
                interW[c]   += sc * w * ist;
                cnt[c]      += ist;
                cntW[c]     += w * ist;
            }
        }
    }

    float acc[NSTAT];
#pragma unroll
    for (int c = 0; c < NCLS; ++c) {
        acc[c]      = sumSoft[c];
        acc[5 + c]  = sumSoftW[c];
        acc[10 + c] = inter[c];
        acc[15 + c] = interW[c];
        acc[20 + c] = cnt[c];
        acc[25 + c] = cntW[c];
    }

    // wave32 tree reduction
#pragma unroll
    for (int i = 0; i < NSTAT; ++i) {
#pragma unroll
        for (int off = 16; off > 0; off >>= 1)
            acc[i] += __shfl_down(acc[i], off, 32);
    }

    __shared__ float lds[TPB / 32][SLOT];
    const int wave = tid >> 5, lane = tid & 31;
    if (lane == 0) {
#pragma unroll
        for (int i = 0; i < NSTAT; ++i) lds[wave][i] = acc[i];
    }
    __syncthreads();
    if (tid < SLOT) {
        float s = 0.0f;
        if (tid < NSTAT) {
#pragma unroll
            for (int wv = 0; wv < TPB / 32; ++wv) s += lds[wv][tid];
        }
        ws[(size_t)blockIdx.x * SLOT + tid] = s;   // non-atomic, deterministic
    }
}

// ---------------------------------------------------------------------------
// Kernel 2: single wave32. Lane = image. Fold block partials, compute dice per
// (image,class), then sum the 160 combined dice values with three chained
// V_WMMA_F32_16X16X4_F32 ops (A = ones -> D accumulates column sums of B).
// ---------------------------------------------------------------------------
__global__ __launch_bounds__(32) void bd_final(const float* __restrict__ ws,
                                               float* __restrict__ out)
{
    const int lane = threadIdx.x;          // 0..31, full wave, EXEC all ones

    float st[NSTAT];
#pragma unroll
    for (int i = 0; i < NSTAT; ++i) st[i] = 0.0f;
    for (int blk = 0; blk < BPI; ++blk) {
        const float* p = ws + ((size_t)lane * BPI + blk) * SLOT;
#pragma unroll
        for (int i = 0; i < NSTAT; ++i) st[i] += p[i];
    }

    float comb[NCLS];
#pragma unroll
    for (int c = 0; c < NCLS; ++c) {
        const float dice_std = (2.0f * st[10 + c] + EPSF) / (st[c]     + st[20 + c] + EPSF);
        const float dice_b   = (2.0f * st[15 + c] + EPSF) / (st[5 + c] + st[25 + c] + EPSF);
        comb[c] = 0.2f * dice_std + 0.8f * dice_b;
    }

    // Sum 160 values: B operands carry 64 values per WMMA; with A = ones(16x4),
    // D[m][n] = sum_k B[k][n]; accumulating C chains the total across WMMAs.
    v2f onesA; onesA.x = 1.0f; onesA.y = 1.0f;
    v2f b0; b0.x = comb[0]; b0.y = comb[1];
    v2f b1; b1.x = comb[2]; b1.y = comb[3];
    v2f b2; b2.x = comb[4]; b2.y = 0.0f;
    v8f d = {};
    d = __builtin_amdgcn_wmma_f32_16x16x4_f32(false, onesA, false, b0, (short)0, d, false, false);
    d = __builtin_amdgcn_wmma_f32_16x16x4_f32(false, onesA, false, b1, (short)0, d, false, false);
    d = __builtin_amdgcn_wmma_f32_16x16x4_f32(false, onesA, false, b2, (short)0, d, false, false);

    // d[0] lane n (n=0..15) holds column-sum n (rows identical); sum lanes 0..15.
    float r = d[0];
    r += __shfl_xor(r, 1, 32);
    r += __shfl_xor(r, 2, 32);
    r += __shfl_xor(r, 4, 32);
    r += __shfl_xor(r, 8, 32);

    if (lane == 0)
        out[0] = 1.0f - r * (1.0f / 160.0f);
}

// ---------------------------------------------------------------------------
extern "C" void kernel_launch(void* const* d_in, const int* in_sizes, int n_in,
                              void* d_out, int out_size, void* d_ws, size_t ws_size,
                              hipStream_t stream)
{
    const float* pred = (const float*)d_in[0];
    const int*   tgt  = (const int*)d_in[1];
    float*       out  = (float*)d_out;
    float*       ws   = (float*)d_ws;          // needs 1024 * 32 * 4 = 128 KB

    (void)in_sizes; (void)n_in; (void)out_size; (void)ws_size;

    bd_partial<<<BIMG * BPI, TPB, 0, stream>>>(pred, tgt, ws);
    bd_final<<<1, 32, 0, stream>>>(ws, out);
}